// SpatialAttention3D_39505109189258
// MI455X (gfx1250) — compile-verified
//
#include <hip/hip_runtime.h>
#include <hip/hip_bf16.h>
#include <math.h>

typedef float v2f __attribute__((ext_vector_type(2)));
typedef float v8f __attribute__((ext_vector_type(8)));

#define DDIM  24
#define DHW   (24 * 24 * 24)   // 13824
#define CCH   128
#define NH    4
#define HD    32
#define BATCH 2

#define LDSROW 68              // 64 cols + 4 pad floats (272B, 16B-aligned)

// ---------------------------------------------------------------------------
// GEMM: Y[b][m][n] = sum_k W[m][k] * X[b][k][n] + bias[m]
// Block (8 waves) computes a 128x64 tile. B (16K x 64N chunk) is staged into
// LDS with double-buffered global_load_async_to_lds_b128 (ASYNCcnt), shared
// by all 8 waves. Each wave: one 16-row M-subtile x four 16-col N-subtiles,
// chained V_WMMA_F32_16X16X4_F32.
// A layout (ISA 7.12.2): lane=(half,idx), vgpr r = A[M=idx][K=2*half+r].
// B: vgpr r = B[K=2*half+r][N=idx].  C/D: vgpr r = C[M=r+8*half][N=idx].
// ---------------------------------------------------------------------------
__global__ __launch_bounds__(256) void wmma_gemm_f32(
    const float* __restrict__ Wmat, const float* __restrict__ bias,
    const float* __restrict__ X, float* __restrict__ Y,
    int M, int K, int N) {
  __shared__ __align__(16) float sb[2][16 * LDSROW];

  const int tid  = threadIdx.x;
  const int lane = tid & 31;
  const int wave = tid >> 5;              // 8 waves -> 8 M-subtiles
  const int half = lane >> 4;
  const int idx  = lane & 15;
  const int nblk = blockIdx.x * 64;
  const int m0   = blockIdx.y * 128 + wave * 16;
  const int b    = blockIdx.z;

  const float* xb = X + (size_t)b * K * N + nblk;
  float*       y  = Y + (size_t)b * M * N;

  // staging geometry: 256 threads move a 16x64 float chunk, 16B per thread
  const int srow = tid >> 4;              // 0..15 (K row within chunk)
  const int scol = (tid & 15) << 2;       // 0..60 (N col, float4 granule)
  const float* gstage = xb + (size_t)srow * N + scol;

  v8f acc[4] = {};

  // prime the pipeline: stage chunk 0 into buffer 0
  {
    unsigned lds = (unsigned)(size_t)&sb[0][srow * LDSROW + scol];
    asm volatile("global_load_async_to_lds_b128 %0, %1, off"
                 :: "v"(lds), "v"(gstage) : "memory");
  }

  const float* ap = Wmat + (size_t)(m0 + idx) * K + 2 * half;

  for (int kc = 0; kc < K; kc += 16) {
    const int cur = (kc >> 4) & 1;
    if (kc + 16 < K) {
      // stage next chunk into the other buffer, then wait for current chunk
      unsigned lds = (unsigned)(size_t)&sb[cur ^ 1][srow * LDSROW + scol];
      const float* g = gstage + (size_t)(kc + 16) * N;
      asm volatile("global_load_async_to_lds_b128 %0, %1, off"
                   :: "v"(lds), "v"(g) : "memory");
      asm volatile("s_wait_asynccnt 0x1" ::: "memory");
    } else {
      asm volatile("s_wait_asynccnt 0x0" ::: "memory");
    }
    __syncthreads();

    // A fragments for the whole 16-K chunk (overlap with LDS-fed WMMAs)
    v2f a[4];
#pragma unroll
    for (int st = 0; st < 4; ++st) {
      a[st].x = ap[4 * st];
      a[st].y = ap[4 * st + 1];
    }
    ap += 16;

    const float* s = &sb[cur][0];
#pragma unroll
    for (int st = 0; st < 4; ++st) {
      const int r0 = 4 * st + 2 * half;
#pragma unroll
      for (int j = 0; j < 4; ++j) {
        v2f bm;
        bm.x = s[r0 * LDSROW + 16 * j + idx];
        bm.y = s[(r0 + 1) * LDSROW + 16 * j + idx];
        acc[j] = __builtin_amdgcn_wmma_f32_16x16x4_f32(
            /*neg_a=*/false, a[st], /*neg_b=*/false, bm,
            /*c_mod=*/(short)0, acc[j], /*reuse_a=*/false, /*reuse_b=*/false);
      }
    }
    __syncthreads();   // protect buffer before it is restaged
  }

#pragma unroll
  for (int j = 0; j < 4; ++j)
#pragma unroll
    for (int r = 0; r < 8; ++r) {
      const int m = m0 + r + 8 * half;
      y[(size_t)m * N + nblk + 16 * j + idx] = acc[j][r] + bias[m];
    }
}

// ---------------------------------------------------------------------------
// Windowed 3x3x3 attention, one thread per (batch, head, voxel).
// qkv layout: [B, 3*C, DHW] with q = ch [0,128), k = [128,256), v = [256,384).
// score_w = (q . (k_w + pe_w)) * hd^-0.5 ; OOB windows: k=0 (pe still counts),
// v contribution zero.
// ---------------------------------------------------------------------------
__global__ __launch_bounds__(256) void attn3d(
    const float* __restrict__ qkv, const float* __restrict__ pe,
    float* __restrict__ out) {
  const int tid = blockIdx.x * blockDim.x + threadIdx.x;
  if (tid >= BATCH * NH * DHW) return;

  const int s = tid % DHW;
  const int n = (tid / DHW) % NH;
  const int b = tid / (DHW * NH);
  const int w = s % DDIM;
  const int h = (s / DDIM) % DDIM;
  const int d = s / (DDIM * DDIM);

  const size_t bstride = (size_t)(3 * CCH) * DHW;
  const float* qp = qkv + (size_t)b * bstride + (size_t)(n * HD) * DHW + s;
  const float* kp = qkv + (size_t)b * bstride + (size_t)(CCH + n * HD) * DHW;
  const float* vp = qkv + (size_t)b * bstride + (size_t)(2 * CCH + n * HD) * DHW;

  float qr[HD];
#pragma unroll
  for (int c = 0; c < HD; ++c) qr[c] = qp[(size_t)c * DHW];

  float sc[27];
  int   s2a[27];

  // pos-embed contribution: uniform addresses -> scalar (SMEM) loads
#pragma unroll
  for (int ww = 0; ww < 27; ++ww) sc[ww] = 0.0f;
#pragma unroll 8
  for (int c = 0; c < HD; ++c) {
    const float qc = qr[c];
#pragma unroll
    for (int ww = 0; ww < 27; ++ww) sc[ww] += qc * pe[c * 27 + ww];
  }

  const float scale = 0.17677669529663687f;  // 32^-0.5
  float mx = -3.0e38f;
#pragma unroll
  for (int dz = -1; dz <= 1; ++dz)
#pragma unroll
  for (int dy = -1; dy <= 1; ++dy)
#pragma unroll
  for (int dx = -1; dx <= 1; ++dx) {
    const int wi = (dz + 1) * 9 + (dy + 1) * 3 + (dx + 1);
    const int zd = d + dz, yh = h + dy, xw = w + dx;
    const bool inb = ((unsigned)zd < (unsigned)DDIM) &
                     ((unsigned)yh < (unsigned)DDIM) &
                     ((unsigned)xw < (unsigned)DDIM);
    const int s2 = (zd * DDIM + yh) * DDIM + xw;
    s2a[wi] = inb ? s2 : -1;
    if (inb) {
      float acc = 0.0f;
#pragma unroll
      for (int c = 0; c < HD; ++c)
        acc += qr[c] * kp[(size_t)c * DHW + s2];
      sc[wi] += acc;
    }
    sc[wi] *= scale;
    mx = fmaxf(mx, sc[wi]);
  }

  // softmax over the 27 window positions
  float sum = 0.0f;
#pragma unroll
  for (int ww = 0; ww < 27; ++ww) {
    const float e = __expf(sc[ww] - mx);
    sc[ww] = e;
    sum += e;
  }
  const float inv = 1.0f / sum;
#pragma unroll
  for (int ww = 0; ww < 27; ++ww) {
    sc[ww] *= inv;
    if (s2a[ww] < 0) { sc[ww] = 0.0f; s2a[ww] = s; }  // zero weight, safe addr
  }

  float* op = out + ((size_t)(b * NH + n) * HD) * DHW + s;
#pragma unroll 8
  for (int c = 0; c < HD; ++c) {
    const float* vc = vp + (size_t)c * DHW;
    float acc = 0.0f;
#pragma unroll
    for (int ww = 0; ww < 27; ++ww)
      acc += sc[ww] * vc[s2a[ww]];
    op[(size_t)c * DHW] = acc;
  }
}

// ---------------------------------------------------------------------------
extern "C" void kernel_launch(void* const* d_in, const int* in_sizes, int n_in,
                              void* d_out, int out_size, void* d_ws, size_t ws_size,
                              hipStream_t stream) {
  const float* x      = (const float*)d_in[0];  // [B,128,24,24,24]
  const float* qkv_w  = (const float*)d_in[1];  // [384,128]
  const float* qkv_b  = (const float*)d_in[2];  // [384]
  const float* proj_w = (const float*)d_in[3];  // [128,128]
  const float* proj_b = (const float*)d_in[4];  // [128]
  const float* pe     = (const float*)d_in[5];  // [32,3,3,3]

  float* qkv = (float*)d_ws;                              // B*384*DHW floats
  float* att = qkv + (size_t)BATCH * 3 * CCH * DHW;       // B*128*DHW floats
  float* outp = (float*)d_out;

  // 1) QKV projection: [384x128] x [128x13824] per batch (3x 128-row blocks)
  {
    dim3 blk(256);
    dim3 grd(DHW / 64, (3 * CCH) / 128, BATCH);
    wmma_gemm_f32<<<grd, blk, 0, stream>>>(qkv_w, qkv_b, x, qkv,
                                           3 * CCH, CCH, DHW);
  }

  // 2) windowed attention
  {
    const int total = BATCH * NH * DHW;
    attn3d<<<(total + 255) / 256, 256, 0, stream>>>(qkv, pe, att);
  }

  // 3) output projection: [128x128] x [128x13824] per batch
  {
    dim3 blk(256);
    dim3 grd(DHW / 64, CCH / 128, BATCH);
    wmma_gemm_f32<<<grd, blk, 0, stream>>>(proj_w, proj_b, att, outp,
                                           CCH, CCH, DHW);
  }
}